// ResonantAttention_75892072121101
// MI455X (gfx1250) — compile-verified
//
#include <hip/hip_runtime.h>
#include <math.h>

// ---------------------------------------------------------------------------
// ResonantAttention for MI455X (gfx1250, wave32).
// B=8, H=4096, D=1024, NH=16, DH=64 (derived from in_sizes at launch).
// Memory-bound: two passes over 128MB cached_states (2nd pass L2-resident).
// WMMA f32 16x16x4 used for the output projection GEMM (M=8 padded to 16,
// padding provided by a zero-filled 16xD context tile -> branch-free loop).
// ---------------------------------------------------------------------------

typedef float v2f __attribute__((ext_vector_type(2)));
typedef float v8f __attribute__((ext_vector_type(8)));

#define PHI_F 1.6180339887498949f

// ---------------------------------------------------------------------------
// Kernel 1: scores[b,n,h] = (sum_d cos(tq)*cos(tk) + sin(tq)*sin(tk)) / scale
// One block handles (b, 32 consecutive h rows). Query sin/cos + key params are
// staged in LDS once per block. Each thread owns 4 consecutive columns (float4
// coalesced row loads); 16 threads cover one head (DH=64); shfl reduction.
// ---------------------------------------------------------------------------
template <int ROWS>
__global__ void __launch_bounds__(256)
scores_kernel(const float* __restrict__ x, const float* __restrict__ cached,
              const float* __restrict__ t,
              const float* __restrict__ wq, const float* __restrict__ bq,
              const float* __restrict__ wk, const float* __restrict__ bk,
              float* __restrict__ scores,
              int H, int D, int NH, float inv_scale)
{
    __shared__ float s_cosq[1024];
    __shared__ float s_sinq[1024];
    __shared__ float s_invk[1024];
    __shared__ float s_bk[1024];

    const int b   = blockIdx.x;
    const int h0  = blockIdx.y * ROWS;
    const int tid = threadIdx.x;
    const int DH  = D / NH;

    const float tphi = t[b] * PHI_F;
    for (int j = tid; j < D; j += 256) {
        const float iq  = 1.0f / (1.0f + fabsf(wq[j]));
        const float thq = x[(size_t)b * D + j] * iq + bq[j] + tphi;
        float sq, cq;
        __sincosf(thq, &sq, &cq);
        s_cosq[j] = cq;
        s_sinq[j] = sq;
        s_invk[j] = 1.0f / (1.0f + fabsf(wk[j]));
        s_bk[j]   = bk[j];
    }
    __syncthreads();

    const int j0 = tid * 4;        // 256 threads * 4 = 1024 columns
    const int n  = j0 / DH;        // head index (16 threads per head)
    const float i0 = s_invk[j0+0], i1 = s_invk[j0+1], i2 = s_invk[j0+2], i3 = s_invk[j0+3];
    const float k0 = s_bk[j0+0],   k1 = s_bk[j0+1],   k2 = s_bk[j0+2],   k3 = s_bk[j0+3];
    const float c0 = s_cosq[j0+0], c1 = s_cosq[j0+1], c2 = s_cosq[j0+2], c3 = s_cosq[j0+3];
    const float q0 = s_sinq[j0+0], q1 = s_sinq[j0+1], q2 = s_sinq[j0+2], q3 = s_sinq[j0+3];

    const float* __restrict__ base = cached + ((size_t)b * H + h0) * D + j0;

    for (int r = 0; r < ROWS; ++r) {
        const float4 kv = *(const float4*)(base + (size_t)r * D);
        float s, c, acc;
        __sincosf(kv.x * i0 + k0, &s, &c); acc  = c * c0 + s * q0;
        __sincosf(kv.y * i1 + k1, &s, &c); acc += c * c1 + s * q1;
        __sincosf(kv.z * i2 + k2, &s, &c); acc += c * c2 + s * q2;
        __sincosf(kv.w * i3 + k3, &s, &c); acc += c * c3 + s * q3;
        // reduce across the 16 lanes of this head segment
        acc += __shfl_down(acc, 8, 16);
        acc += __shfl_down(acc, 4, 16);
        acc += __shfl_down(acc, 2, 16);
        acc += __shfl_down(acc, 1, 16);
        if ((tid & 15) == 0)
            scores[((size_t)b * NH + n) * H + (h0 + r)] = acc * inv_scale;
    }
}

// ---------------------------------------------------------------------------
// Kernel 2: softmax over h (row length H) per (b,n). 256 threads = 8 waves.
// ---------------------------------------------------------------------------
__global__ void __launch_bounds__(256)
softmax_kernel(float* __restrict__ scores, int H)
{
    float* __restrict__ row = scores + (size_t)blockIdx.x * H;
    const int tid = threadIdx.x;
    __shared__ float red[8];

    float m = -3.402823466e38f;
    for (int i = tid; i < H; i += 256) m = fmaxf(m, row[i]);
#pragma unroll
    for (int o = 16; o >= 1; o >>= 1) m = fmaxf(m, __shfl_xor(m, o, 32));
    if ((tid & 31) == 0) red[tid >> 5] = m;
    __syncthreads();
    if (tid < 8) {
        float v = red[tid];
#pragma unroll
        for (int o = 4; o >= 1; o >>= 1) v = fmaxf(v, __shfl_xor(v, o, 8));
        if (tid == 0) red[0] = v;
    }
    __syncthreads();
    m = red[0];
    __syncthreads();

    float sum = 0.0f;
    for (int i = tid; i < H; i += 256) {
        const float e = __expf(row[i] - m);
        row[i] = e;
        sum += e;
    }
#pragma unroll
    for (int o = 16; o >= 1; o >>= 1) sum += __shfl_xor(sum, o, 32);
    if ((tid & 31) == 0) red[tid >> 5] = sum;
    __syncthreads();
    if (tid < 8) {
        float v = red[tid];
#pragma unroll
        for (int o = 4; o >= 1; o >>= 1) v += __shfl_xor(v, o, 8);
        if (tid == 0) red[0] = v;
    }
    __syncthreads();
    const float inv = 1.0f / red[0];
    for (int i = tid; i < H; i += 256) row[i] *= inv;
}

// ---------------------------------------------------------------------------
// Kernel 3: W2[b,h] = sum_n weights[b,n,h]  (head collapse before context GEMV)
// ---------------------------------------------------------------------------
__global__ void __launch_bounds__(256)
w2_kernel(const float* __restrict__ weights, float* __restrict__ W2,
          int B, int H, int NH)
{
    const int idx = blockIdx.x * 256 + threadIdx.x;
    if (idx >= B * H) return;
    const int b = idx / H, h = idx - b * H;
    const float* __restrict__ p = weights + (size_t)b * NH * H + h;
    float s = 0.0f;
#pragma unroll 4
    for (int n = 0; n < NH; ++n) s += p[(size_t)n * H];
    W2[idx] = s;
}

__global__ void zero_kernel(float* __restrict__ p, int n)
{
    const int i = blockIdx.x * 256 + threadIdx.x;
    if (i < n) p[i] = 0.0f;
}

// ---------------------------------------------------------------------------
// Kernel 4: context[b,m] += sum_{h in seg} W2[b,h] * cached[b,h,m]
// Split-K over 16 h-segments, f32 atomics. Second pass over cached_states —
// expected L2-resident (128MB < 192MB). Prefetch hints on the stream.
// ---------------------------------------------------------------------------
__global__ void __launch_bounds__(256)
context_kernel(const float* __restrict__ cached, const float* __restrict__ W2,
               float* __restrict__ context, int H, int D, int nseg)
{
    const int b   = blockIdx.x;
    const int m   = blockIdx.y * 256 + threadIdx.x;
    const int hseg = H / nseg;
    const int h0   = blockIdx.z * hseg;

    const float* __restrict__ cs = cached + ((size_t)b * H + h0) * D + m;
    const float* __restrict__ w  = W2 + (size_t)b * H + h0;

    float acc = 0.0f;
    for (int h = 0; h < hseg; ++h) {
        if (((threadIdx.x & 31) == 0) && (h + 8 < hseg))
            __builtin_prefetch(cs + (size_t)(h + 8) * D, 0, 3);
        acc += w[h] * cs[(size_t)h * D];
    }
    atomicAdd(&context[(size_t)b * D + m], acc);
}

// ---------------------------------------------------------------------------
// Kernel 5: out[b,j] = x[b,j] + sum_m context16[b,m] * out_w[j,m]
// GEMM (M=16 incl. zero padding rows 8..15, N=1024, K=1024) on
// V_WMMA_F32_16X16X4_F32. One wave per 16-column tile. context16 is a
// zero-padded 16xD tile so all fragment loads are branch-free b64 loads.
// Fragment layouts per ISA 7.12.2:
//   A 16x4 f32 : lanes 0-15 M=0-15 K={0,1}; lanes 16-31 M=0-15 K={2,3}
//   B 4x16 f32 : VGPR0 rows K=0|K=2, VGPR1 rows K=1|K=3 striped over lanes
//   C/D        : VGPR r -> (M=r, N=lane) | (M=8+r, N=lane-16)
// K-loop is software-pipelined (double-buffered fragments) so loads for k+4
// issue before the WMMA for k waits.
// ---------------------------------------------------------------------------
__global__ void __launch_bounds__(32)
proj_wmma_kernel(const float* __restrict__ context16, const float* __restrict__ out_w,
                 const float* __restrict__ x, float* __restrict__ out,
                 int D, int Brows)
{
    const int lane = threadIdx.x;       // 0..31, full wave, EXEC all ones
    const int j0   = blockIdx.x * 16;   // output column tile
    const int half = lane >> 4;         // K-pair selector per ISA layout
    const int l15  = lane & 15;

    // A row M=l15 (rows >= Brows are pre-zeroed); B column N=l15.
    const float* __restrict__ arow = context16 + (size_t)l15 * D + half * 2;
    const float* __restrict__ bcol = out_w + (size_t)(j0 + l15) * D + half * 2;

    v8f c = {0.f, 0.f, 0.f, 0.f, 0.f, 0.f, 0.f, 0.f};

    v2f a0 = *(const v2f*)(arow);
    v2f b0 = *(const v2f*)(bcol);

#pragma unroll 8
    for (int k = 4; k < D; k += 4) {
        const v2f a1 = *(const v2f*)(arow + k);     // prefetch next fragments
        const v2f b1 = *(const v2f*)(bcol + k);
        c = __builtin_amdgcn_wmma_f32_16x16x4_f32(
                false, a0, false, b0, (short)0, c, false, false);
        a0 = a1;
        b0 = b1;
    }
    c = __builtin_amdgcn_wmma_f32_16x16x4_f32(
            false, a0, false, b0, (short)0, c, false, false);

    if (half == 0) {
#pragma unroll
        for (int r = 0; r < 8; ++r) {
            if (r < Brows) {
                const size_t o = (size_t)r * D + j0 + l15;
                out[o] = x[o] + c[r];
            }
        }
    }
}

// ---------------------------------------------------------------------------
extern "C" void kernel_launch(void* const* d_in, const int* in_sizes, int n_in,
                              void* d_out, int out_size, void* d_ws, size_t ws_size,
                              hipStream_t stream)
{
    const float* x      = (const float*)d_in[0];
    const float* cached = (const float*)d_in[1];
    const float* t      = (const float*)d_in[2];
    const float* wq     = (const float*)d_in[3];
    const float* bq     = (const float*)d_in[4];
    const float* wk     = (const float*)d_in[5];
    const float* bk     = (const float*)d_in[6];
    const float* out_w  = (const float*)d_in[7];
    float*       out    = (float*)d_out;

    const int B  = in_sizes[2];                 // 8
    const int D  = in_sizes[0] / B;             // 1024
    const int H  = in_sizes[1] / in_sizes[0];   // 4096
    const int NH = 16;
    const int DH = D / NH;
    const float inv_scale = 1.0f / sqrtf(2.0f * (float)DH);

    // workspace layout (f32): scores[B*NH*H] | W2[B*H] | context16[16*D]
    float* scores    = (float*)d_ws;
    float* W2        = scores + (size_t)B * NH * H;
    float* context16 = W2 + (size_t)B * H;

    constexpr int ROWS = 32;
    scores_kernel<ROWS><<<dim3(B, H / ROWS), 256, 0, stream>>>(
        x, cached, t, wq, bq, wk, bk, scores, H, D, NH, inv_scale);

    softmax_kernel<<<B * NH, 256, 0, stream>>>(scores, H);

    w2_kernel<<<(B * H + 255) / 256, 256, 0, stream>>>(scores, W2, B, H, NH);

    // zero the full 16xD padded context tile (rows B..15 stay zero -> WMMA pad)
    zero_kernel<<<(16 * D + 255) / 256, 256, 0, stream>>>(context16, 16 * D);

    const int NSEG = 16;
    context_kernel<<<dim3(B, D / 256, NSEG), 256, 0, stream>>>(
        cached, W2, context16, H, D, NSEG);

    proj_wmma_kernel<<<D / 16, 32, 0, stream>>>(context16, out_w, x, out, D, B);
}